// HAN_45775761441404
// MI455X (gfx1250) — compile-verified
//
#include <hip/hip_runtime.h>
#include <math.h>

#define BN_EPS 1e-5f

typedef float v2f __attribute__((ext_vector_type(2)));
typedef float v8f __attribute__((ext_vector_type(8)));
typedef unsigned int v4u __attribute__((ext_vector_type(4)));
typedef int v8i_t __attribute__((ext_vector_type(8)));
typedef int v4i_t __attribute__((ext_vector_type(4)));
typedef __attribute__((address_space(3))) float lds_f;

// ---------------- ordered-uint encoding for float atomic max ----------------
__device__ __forceinline__ unsigned fenc(float f) {
    unsigned u = __float_as_uint(f);
    return (u & 0x80000000u) ? ~u : (u | 0x80000000u);
}
__device__ __forceinline__ float fdec(unsigned e) {
    return (e & 0x80000000u) ? __uint_as_float(e ^ 0x80000000u) : __uint_as_float(~e);
}

// -------- TDM: DMA a 16-row x 128-col f32 tile into LDS with +2-DWORD row pad
// Descriptor per CDNA5 ISA (groups 0/1; groups 2/3 zero).
//   tile_dim0=128, tile_dim1=16, tensor_dim0=128, tensor_dim0_stride=128,
//   tensor_dim1=M, data_size=4B, pad: 2 DWORDs after every 128 DWORDs
//   -> LDS row stride 130 floats == LSTR below.
__device__ __forceinline__ void tdm_load_tile_f32_128(const float* gsrc,
                                                      unsigned lds_byte_off, int M)
{
    unsigned long long ga = (unsigned long long)(size_t)gsrc;
    v4u g0;
    g0[0] = 1u;                                            // count=1, user mode
    g0[1] = lds_byte_off;                                  // lds_addr (bytes)
    g0[2] = (unsigned)(ga & 0xFFFFFFFFu);                  // global_addr[31:0]
    g0[3] = (unsigned)((ga >> 32) & 0x01FFFFFFu) | (2u << 30); // addr[56:32] | type=2
    v8i_t g1;
    g1[0] = (int)((2u << 16)      // data_size = 4B
                | (1u << 20)      // pad_enable
                | (6u << 22)      // pad_interval: 128 DWORDs
                | (1u << 25));    // pad_amount: 2 DWORDs
    g1[1] = (int)(128u << 16);                             // tensor_dim0 = 128 (bits 79:48)
    g1[2] = (int)(((unsigned)M & 0xFFFFu) << 16);          // tensor_dim1 lo (bits 111:80)
    g1[3] = (int)((((unsigned)M >> 16) & 0xFFFFu)          // tensor_dim1 hi
                | (128u << 16));                           // tile_dim0 = 128 (bits 127:112)
    g1[4] = 16;                                            // tile_dim1 = 16
    g1[5] = 128;                                           // tensor_dim0_stride lo
    g1[6] = 0;
    g1[7] = 0;
    v4i_t z4 = {0, 0, 0, 0};
    v8i_t z8 = {0, 0, 0, 0, 0, 0, 0, 0};
    // 6-arg form (clang-23 / therock headers): (g0, g1, g2, g3, g4, cpol)
    __builtin_amdgcn_tensor_load_to_lds(g0, g1, z4, z4, z8, 0);
}

// ---------------- WMMA f32 GEMM: D[M,N] = op(A)[M,K] @ W[K,N] + bias --------
// op(A) = A, or relu(A*scale[k]+shift[k]) when ATRANS (fused BN+ReLU).
// EPI==0: store D.  EPI==1: colsum[col] += sum_rows tanh(d) (semantic attn).
// A-tile staging: TDM async-tensor DMA when !ATRANS && K==128, else manual.
template<int KDIM, int NCOLS, bool ATRANS, int EPI>
__global__ __launch_bounds__(32*(NCOLS/16))
void wmma_gemm_kernel(const float* __restrict__ A,
                      const float* __restrict__ W,
                      const float* __restrict__ bias,
                      const float* __restrict__ a_scale,
                      const float* __restrict__ a_shift,
                      float* __restrict__ D,
                      float* __restrict__ colsum,
                      int M)
{
    constexpr int NWAVES = NCOLS / 16;
    constexpr int NTHR   = 32 * NWAVES;
    constexpr int LSTR   = KDIM + 2;          // pad to avoid LDS bank conflicts
    __shared__ float As[16 * LSTR];

    const int row0 = blockIdx.x * 16;
    const int tid  = threadIdx.x;
    const int wave = tid >> 5;
    const int lane = tid & 31;
    const int half = lane >> 4;     // 0: K {0,1}, rows 0-7 of D; 1: K {2,3}, rows 8-15
    const int l15  = lane & 15;
    const int col  = wave * 16 + l15;

    // warm this wave's B-column slice of W while the A tile is staged
    #pragma unroll
    for (int kk = 0; kk < KDIM; kk += 32)
        __builtin_prefetch(&W[(size_t)kk * NCOLS + col], 0, 1);

    if (!ATRANS && KDIM == 128) {
        // async-tensor DMA of the whole 16x128 tile (wave 0 issues; TDM ignores EXEC)
        if (tid < 32) {
            unsigned lds_off = (unsigned)(unsigned long long)(lds_f*)As;
            tdm_load_tile_f32_128(A + (size_t)row0 * KDIM, lds_off, M);
            __builtin_amdgcn_s_wait_tensorcnt((short)0);
        }
    } else {
        // cooperative coalesced stage with fused BN+ReLU
        for (int i = tid; i < 16 * KDIM; i += NTHR) {
            int r = i / KDIM, k = i - r * KDIM;
            float v = A[(size_t)(row0 + r) * KDIM + k];
            if (ATRANS) v = fmaxf(v * a_scale[k] + a_shift[k], 0.0f);
            As[r * LSTR + k] = v;
        }
    }
    __syncthreads();

    v8f acc = {};
    #pragma unroll 4
    for (int k0 = 0; k0 < KDIM; k0 += 4) {
        const int ka = k0 + half * 2;
        v2f a, b;
        a.x = As[l15 * LSTR + ka];
        a.y = As[l15 * LSTR + ka + 1];
        b.x = W[(size_t)ka * NCOLS + col];
        b.y = W[(size_t)(ka + 1) * NCOLS + col];
        acc = __builtin_amdgcn_wmma_f32_16x16x4_f32(false, a, false, b,
                                                    (short)0, acc, false, false);
    }
    const float bv = bias ? bias[col] : 0.0f;

    if (EPI == 0) {
        const int rbase = half * 8;
        #pragma unroll
        for (int r = 0; r < 8; ++r)
            D[(size_t)(row0 + rbase + r) * NCOLS + col] = acc[r] + bv;
    } else {
        float s = 0.0f;
        #pragma unroll
        for (int r = 0; r < 8; ++r) s += tanhf(acc[r] + bv);
        atomicAdd(&colsum[col], s);
    }
}

// ------------- per-node attention logit dot products (6 arrays) -------------
__global__ void alpha_kernel(const float* __restrict__ ha, const float* __restrict__ hp,
                             const float* __restrict__ as_ap, const float* __restrict__ ad_ap,
                             const float* __restrict__ as_pa, const float* __restrict__ ad_pa,
                             const float* __restrict__ as_pp, const float* __restrict__ ad_pp,
                             float* __restrict__ s_ap, float* __restrict__ d_ap,
                             float* __restrict__ s_pa, float* __restrict__ d_pa,
                             float* __restrict__ s_pp, float* __restrict__ d_pp, int Nn)
{
    int i = blockIdx.x * blockDim.x + threadIdx.x;
    if (i >= Nn * 8) return;
    int n = i >> 3, h = i & 7;
    const float* va = ha + (size_t)n * 128 + h * 16;
    const float* vp = hp + (size_t)n * 128 + h * 16;
    float r0 = 0, r1 = 0, r2 = 0, r3 = 0, r4 = 0, r5 = 0;
    #pragma unroll
    for (int d = 0; d < 16; ++d) {
        float xa = va[d], xp = vp[d];
        int ai = h * 16 + d;
        r0 += xa * as_ap[ai]; r1 += xp * ad_ap[ai];
        r2 += xp * as_pa[ai]; r3 += xa * ad_pa[ai];
        r4 += xp * as_pp[ai]; r5 += xp * ad_pp[ai];
    }
    s_ap[i] = r0; d_ap[i] = r1; s_pa[i] = r2;
    d_pa[i] = r3; s_pp[i] = r4; d_pp[i] = r5;
}

// ----------------- edge pass 1: leaky-relu logits + segment max -------------
__global__ void edge_max_kernel(const int* __restrict__ ei, int E,
                                const float* __restrict__ sarr, const float* __restrict__ darr,
                                float* __restrict__ ebuf, unsigned* __restrict__ mx)
{
    int i = blockIdx.x * blockDim.x + threadIdx.x;
    if (i >= E * 8) return;
    int e = i >> 3, h = i & 7;
    int src = ei[e], dst = ei[E + e];
    float l = sarr[(size_t)src * 8 + h] + darr[(size_t)dst * 8 + h];
    l = (l > 0.0f) ? l : 0.2f * l;                 // leaky_relu(0.2)
    ebuf[i] = l;
    atomicMax(&mx[(size_t)dst * 8 + h], fenc(l));
}

// ----------------- edge pass 2: exp(l - max) + segment sum ------------------
__global__ void edge_expsum_kernel(const int* __restrict__ ei, int E,
                                   const unsigned* __restrict__ mx,
                                   float* __restrict__ ebuf, float* __restrict__ ssum)
{
    int i = blockIdx.x * blockDim.x + threadIdx.x;
    if (i >= E * 8) return;
    int e = i >> 3, h = i & 7;
    int dst = ei[E + e];
    float m  = fdec(mx[(size_t)dst * 8 + h]);
    float ex = __expf(ebuf[i] - m);
    ebuf[i] = ex;
    atomicAdd(&ssum[(size_t)dst * 8 + h], ex);
}

// ----------------- edge pass 3: scatter h_src * alpha into dst --------------
__global__ void edge_agg_kernel(const int* __restrict__ ei, int E,
                                const float* __restrict__ hsrc,
                                const float* __restrict__ ebuf,
                                const float* __restrict__ ssum,
                                float* __restrict__ outacc)
{
    int gw   = (blockIdx.x * blockDim.x + threadIdx.x) >> 5;  // one wave32 per edge
    int lane = threadIdx.x & 31;
    if (gw >= E) return;
    int src = ei[gw], dst = ei[E + gw];
    #pragma unroll
    for (int i = 0; i < 4; ++i) {
        int c = lane + 32 * i;
        int h = c >> 4;
        float coef = ebuf[(size_t)gw * 8 + h] / (ssum[(size_t)dst * 8 + h] + 1e-16f);
        atomicAdd(&outacc[(size_t)dst * 128 + c], hsrc[(size_t)src * 128 + c] * coef);
    }
}

// --------------------------- misc elementwise -------------------------------
__global__ void relu_inplace_kernel(float* __restrict__ x, long n) {
    long i = (long)blockIdx.x * blockDim.x + threadIdx.x;
    if (i < n) x[i] = fmaxf(x[i], 0.0f);
}

__global__ void sem_score_kernel(const float* __restrict__ q,
                                 const float* __restrict__ colsum,
                                 float invN, float* __restrict__ attn)
{
    __shared__ float red0[128], red1[128];
    int t = threadIdx.x;  // 128 threads
    red0[t] = q[t] * colsum[t] * invN;
    red1[t] = q[t] * colsum[128 + t] * invN;
    __syncthreads();
    for (int s = 64; s > 0; s >>= 1) {
        if (t < s) { red0[t] += red0[t + s]; red1[t] += red1[t + s]; }
        __syncthreads();
    }
    if (t == 0) {
        float s0 = red0[0], s1 = red1[0];
        float m = fmaxf(s0, s1);
        float e0 = __expf(s0 - m), e1 = __expf(s1 - m);
        attn[0] = e0 / (e0 + e1);
        attn[1] = e1 / (e0 + e1);
    }
}

__global__ void xp_combine_kernel(const float* __restrict__ a, const float* __restrict__ b,
                                  const float* __restrict__ attn, float* __restrict__ o, long n)
{
    long i = (long)blockIdx.x * blockDim.x + threadIdx.x;
    if (i < n) o[i] = attn[0] * a[i] + attn[1] * b[i];
}

// --------------------------- batchnorm helpers ------------------------------
template<int CCH>
__global__ void stats_kernel(const float* __restrict__ X, int M,
                             float* __restrict__ gsum, float* __restrict__ gsumsq)
{
    int t = threadIdx.x;                // CCH threads, one per channel
    int row0 = blockIdx.x * 64;
    float s = 0.0f, ss = 0.0f;
    for (int r = 0; r < 64; ++r) {
        int row = row0 + r;
        if (row < M) {
            float v = X[(size_t)row * CCH + t];
            s += v; ss += v * v;
        }
    }
    atomicAdd(&gsum[t], s);
    atomicAdd(&gsumsq[t], ss);
}

__global__ void bn_finalize_kernel(const float* __restrict__ gsum, const float* __restrict__ gsumsq,
                                   const float* __restrict__ g, const float* __restrict__ b,
                                   float invM, float* __restrict__ sc, float* __restrict__ sh, int C)
{
    int t = blockIdx.x * blockDim.x + threadIdx.x;
    if (t >= C) return;
    float mean = gsum[t] * invM;
    float var  = fmaxf(gsumsq[t] * invM - mean * mean, 0.0f);
    float s = g[t] * rsqrtf(var + BN_EPS);
    sc[t] = s;
    sh[t] = b[t] - mean * s;
}

// relu(bn1(x)) then AvgPool1d(2): [N,128] -> [N,64]
__global__ void pool_kernel(const float* __restrict__ X, const float* __restrict__ sc,
                            const float* __restrict__ sh, float* __restrict__ P, int M)
{
    int i = blockIdx.x * blockDim.x + threadIdx.x;
    if (i >= M * 64) return;
    int n = i >> 6, j = i & 63;
    float a = fmaxf(X[(size_t)n * 128 + 2 * j]     * sc[2 * j]     + sh[2 * j],     0.0f);
    float b = fmaxf(X[(size_t)n * 128 + 2 * j + 1] * sc[2 * j + 1] + sh[2 * j + 1], 0.0f);
    P[i] = 0.5f * (a + b);
}

__global__ void final_out_kernel(const float* __restrict__ z, const float* __restrict__ sc,
                                 const float* __restrict__ sh, float* __restrict__ out, int M)
{
    int i = blockIdx.x * blockDim.x + threadIdx.x;
    if (i >= M * 64) return;
    int c = i & 63;
    out[i] = fmaxf(z[i] * sc[c] + sh[c], 0.0f);
}

// ============================== driver ======================================
extern "C" void kernel_launch(void* const* d_in, const int* in_sizes, int n_in,
                              void* d_out, int out_size, void* d_ws, size_t ws_size,
                              hipStream_t stream)
{
    const float* x_author = (const float*)d_in[0];
    const float* x_paper  = (const float*)d_in[1];
    const float* lin_a_w  = (const float*)d_in[2];
    const float* lin_a_b  = (const float*)d_in[3];
    const float* lin_p_w  = (const float*)d_in[4];
    const float* lin_p_b  = (const float*)d_in[5];
    const float* att_src_ap = (const float*)d_in[6];
    const float* att_dst_ap = (const float*)d_in[7];
    const float* att_src_pa = (const float*)d_in[8];
    const float* att_dst_pa = (const float*)d_in[9];
    const float* att_src_pp = (const float*)d_in[10];
    const float* att_dst_pp = (const float*)d_in[11];
    const float* q       = (const float*)d_in[12];
    const float* klin_w  = (const float*)d_in[13];
    const float* klin_b  = (const float*)d_in[14];
    const float* bn1_g   = (const float*)d_in[15];
    const float* bn1_b   = (const float*)d_in[16];
    const float* bn2_g   = (const float*)d_in[17];
    const float* bn2_b   = (const float*)d_in[18];
    const float* fc1_w   = (const float*)d_in[19];
    const float* fc1_b   = (const float*)d_in[20];
    const float* bn3_g   = (const float*)d_in[21];
    const float* bn3_b   = (const float*)d_in[22];
    const float* fc2_w   = (const float*)d_in[23];
    const float* fc2_b   = (const float*)d_in[24];
    const float* fc3_w   = (const float*)d_in[25];
    const float* fc3_b   = (const float*)d_in[26];
    const float* bn4_g   = (const float*)d_in[27];
    const float* bn4_b   = (const float*)d_in[28];
    const int*   ei_ap   = (const int*)d_in[29];
    const int*   ei_pa   = (const int*)d_in[30];
    const int*   ei_pp   = (const int*)d_in[31];

    const int Nn = in_sizes[0] / 128;      // 50000
    const int E  = in_sizes[29] / 2;       // 400000
    const size_t NC = (size_t)Nn * 128;
    const size_t N8 = (size_t)Nn * 8;

    // ---------------- workspace layout (floats) ----------------
    float* f     = (float*)d_ws;
    float* ha    = f;
    float* hp    = f + NC;
    float* o_ap  = f + 2 * NC;
    float* o_pa  = f + 3 * NC;   // == xa
    float* o_pp  = f + 4 * NC;
    float* s_ap  = f + 5 * NC;
    float* d_ap  = s_ap + N8;
    float* s_pa  = d_ap + N8;
    float* d_pa  = s_pa + N8;
    float* s_pp  = d_pa + N8;
    float* d_pp  = s_pp + N8;
    float* mxf   = d_pp + N8;            // per-(dst,head) running max (encoded)
    float* ssum  = mxf + N8;             // per-(dst,head) exp sum
    float* ebuf  = ssum + N8;            // per-(edge,head) logits -> exp  (E*8)
    float* xp    = ebuf + (size_t)E * 8;
    float* z1    = xp + NC;              // [N,256]
    float* z2    = z1 + 2 * NC;          // [N,256]
    float* pbuf  = z2 + 2 * NC;          // [N,64]
    float* z3    = pbuf + (size_t)Nn * 64; // [N,64]
    float* sm    = z3 + (size_t)Nn * 64; // small scratch
    float* colsum = sm;                  // 256
    float* attn   = sm + 256;            // 2
    float* gsum   = sm + 512;            // 256
    float* gsumsq = sm + 768;            // 256
    float* sc     = sm + 1024;           // 256
    float* sh     = sm + 1280;           // 256

    float* out = (float*)d_out;
    const int rowTiles = Nn / 16;                 // 3125
    const int eh_blocks = (E * 8 + 255) / 256;
    const int agg_blocks = (E * 32 + 255) / 256;  // wave per edge

    // ---------------- 1. input projections (WMMA f32 + TDM staging) ---------
    wmma_gemm_kernel<128, 128, false, 0><<<rowTiles, 256, 0, stream>>>(
        x_author, lin_a_w, lin_a_b, nullptr, nullptr, ha, nullptr, Nn);
    wmma_gemm_kernel<128, 128, false, 0><<<rowTiles, 256, 0, stream>>>(
        x_paper, lin_p_w, lin_p_b, nullptr, nullptr, hp, nullptr, Nn);

    // ---------------- 2. per-node attention logits ----------------
    alpha_kernel<<<(Nn * 8 + 255) / 256, 256, 0, stream>>>(
        ha, hp, att_src_ap, att_dst_ap, att_src_pa, att_dst_pa, att_src_pp, att_dst_pp,
        s_ap, d_ap, s_pa, d_pa, s_pp, d_pp, Nn);

    // ---------------- 3. three relational attentions ----------------
    struct Rel { const float* s; const float* d; const int* ei; const float* hsrc; float* o; };
    Rel rels[3] = {
        { s_ap, d_ap, ei_ap, ha, o_ap },   // author -> paper
        { s_pa, d_pa, ei_pa, hp, o_pa },   // paper  -> author
        { s_pp, d_pp, ei_pp, hp, o_pp },   // paper  -> paper
    };
    for (int r = 0; r < 3; ++r) {
        (void)hipMemsetAsync(mxf,  0, N8 * sizeof(float), stream);
        (void)hipMemsetAsync(ssum, 0, N8 * sizeof(float), stream);
        (void)hipMemsetAsync(rels[r].o, 0, NC * sizeof(float), stream);
        edge_max_kernel<<<eh_blocks, 256, 0, stream>>>(
            rels[r].ei, E, rels[r].s, rels[r].d, ebuf, (unsigned*)mxf);
        edge_expsum_kernel<<<eh_blocks, 256, 0, stream>>>(
            rels[r].ei, E, (const unsigned*)mxf, ebuf, ssum);
        edge_agg_kernel<<<agg_blocks, 256, 0, stream>>>(
            rels[r].ei, E, rels[r].hsrc, ebuf, ssum, rels[r].o);
    }
    // relu on all three outputs (contiguous region)
    relu_inplace_kernel<<<(int)((3 * NC + 255) / 256), 256, 0, stream>>>(o_ap, (long)(3 * NC));

    // ---------------- 4. semantic attention for xp ----------------
    (void)hipMemsetAsync(colsum, 0, 256 * sizeof(float), stream);
    wmma_gemm_kernel<128, 128, false, 1><<<rowTiles, 256, 0, stream>>>(
        o_ap, klin_w, klin_b, nullptr, nullptr, nullptr, colsum, Nn);
    wmma_gemm_kernel<128, 128, false, 1><<<rowTiles, 256, 0, stream>>>(
        o_pp, klin_w, klin_b, nullptr, nullptr, nullptr, colsum + 128, Nn);
    sem_score_kernel<<<1, 128, 0, stream>>>(q, colsum, 1.0f / (float)Nn, attn);
    xp_combine_kernel<<<(int)((NC + 255) / 256), 256, 0, stream>>>(o_ap, o_pp, attn, xp, (long)NC);
    // xa == o_pa (softmax over a single relation is identity)

    // ---------------- 5. MLP on each branch ----------------
    const float invM = 1.0f / (float)Nn;
    auto run_mlp = [&](const float* x, float* obase) {
        // bn1 on [N,128]
        (void)hipMemsetAsync(gsum, 0, 512 * sizeof(float), stream);
        stats_kernel<128><<<(Nn + 63) / 64, 128, 0, stream>>>(x, Nn, gsum, gsumsq);
        bn_finalize_kernel<<<1, 128, 0, stream>>>(gsum, gsumsq, bn1_g, bn1_b, invM, sc, sh, 128);
        // relu(bn1) + avgpool -> p [N,64]
        pool_kernel<<<(Nn * 64 + 255) / 256, 256, 0, stream>>>(x, sc, sh, pbuf, Nn);
        // bn2 on [N,64]
        (void)hipMemsetAsync(gsum, 0, 512 * sizeof(float), stream);
        stats_kernel<64><<<(Nn + 63) / 64, 64, 0, stream>>>(pbuf, Nn, gsum, gsumsq);
        bn_finalize_kernel<<<1, 64, 0, stream>>>(gsum, gsumsq, bn2_g, bn2_b, invM, sc, sh, 64);
        // fc1: relu(bn2(p)) @ [64,256]
        wmma_gemm_kernel<64, 256, true, 0><<<rowTiles, 512, 0, stream>>>(
            pbuf, fc1_w, fc1_b, sc, sh, z1, nullptr, Nn);
        // bn3 on z1 [N,256]
        (void)hipMemsetAsync(gsum, 0, 512 * sizeof(float), stream);
        stats_kernel<256><<<(Nn + 63) / 64, 256, 0, stream>>>(z1, Nn, gsum, gsumsq);
        bn_finalize_kernel<<<1, 256, 0, stream>>>(gsum, gsumsq, bn3_g, bn3_b, invM, sc, sh, 256);
        // fc2: relu(bn3(z1)) @ [256,256]
        wmma_gemm_kernel<256, 256, true, 0><<<rowTiles, 512, 0, stream>>>(
            z1, fc2_w, fc2_b, sc, sh, z2, nullptr, Nn);
        // bn3 again (shared params, fresh batch stats) on z2
        (void)hipMemsetAsync(gsum, 0, 512 * sizeof(float), stream);
        stats_kernel<256><<<(Nn + 63) / 64, 256, 0, stream>>>(z2, Nn, gsum, gsumsq);
        bn_finalize_kernel<<<1, 256, 0, stream>>>(gsum, gsumsq, bn3_g, bn3_b, invM, sc, sh, 256);
        // fc3: relu(bn3(z2)) @ [256,64]
        wmma_gemm_kernel<256, 64, true, 0><<<rowTiles, 128, 0, stream>>>(
            z2, fc3_w, fc3_b, sc, sh, z3, nullptr, Nn);
        // bn4 on z3 [N,64] then relu -> output
        (void)hipMemsetAsync(gsum, 0, 512 * sizeof(float), stream);
        stats_kernel<64><<<(Nn + 63) / 64, 64, 0, stream>>>(z3, Nn, gsum, gsumsq);
        bn_finalize_kernel<<<1, 64, 0, stream>>>(gsum, gsumsq, bn4_g, bn4_b, invM, sc, sh, 64);
        final_out_kernel<<<(Nn * 64 + 255) / 256, 256, 0, stream>>>(z3, sc, sh, obase, Nn);
    };

    run_mlp(o_pa, out);                      // xa branch
    run_mlp(xp,  out + (size_t)Nn * 64);     // xp branch
}